// ReflectionRayTracer_82721070121228
// MI455X (gfx1250) — compile-verified
//
#include <hip/hip_runtime.h>
#include <hip/hip_bf16.h>

// ReflectionRayTracer for MI455X (gfx1250, wave32).
// vo = o @ v.T and vd = d @ v.T are computed as K=4-padded f32 WMMAs
// (V_WMMA_F32_16X16X4_F32, one 16x16 output tile per wave).
// All surface-only terms (plane normal, k, B/D/P, and the three _safe
// reciprocals) are computed per-lane per-column; the per-pair epilogue has a
// single v_rcp (1/vd) and ~25 VALU ops.

#define N_RAYS 8192
#define N_SURF 2048
#define EPSF   1e-18f

typedef float v2f __attribute__((ext_vector_type(2)));
typedef float v8f __attribute__((ext_vector_type(8)));

__device__ __forceinline__ float safe_rcp(float x) {
    return __builtin_amdgcn_rcpf(x == 0.0f ? EPSF : x);
}

__global__ __launch_bounds__(256) void
ReflectionRayTracer_kernel(const float* __restrict__ o,
                           const float* __restrict__ d,
                           const float* __restrict__ V,
                           float* __restrict__ out) {
    const int lane   = threadIdx.x & 31;
    const int wid    = threadIdx.x >> 5;
    const int gw     = blockIdx.x * 8 + wid;          // 65536 waves total
    const int surfT  = gw & (N_SURF / 16 - 1);        // 128 surface tiles
    const int rayT   = gw >> 7;                       // 512 ray tiles
    const int rayBase  = rayT * 16;
    const int surfBase = surfT * 16;
    const int laneLo = lane & 15;
    const int laneHi = lane >> 4;                     // 0: K=0,1  1: K=2,3

    // ---------------- per-surface constants for this lane's column ----------
    const int col = surfBase + laneLo;
    const float* __restrict__ q = V + (size_t)col * 12;
    const float a0x = q[0],  a0y = q[1],  a0z = q[2];   // V0 (= "a")
    const float b1x = q[3],  b1y = q[4],  b1z = q[5];   // V1 (= "b")
    const float c2x = q[6],  c2y = q[7],  c2z = q[8];   // V2 (= "c")
    const float p3x = q[9],  p3y = q[10], p3z = q[11];  // V3

    // normal = normalize(cross(V1-V0, V2-V0))
    const float e1x = b1x - a0x, e1y = b1y - a0y, e1z = b1z - a0z;
    const float e2x = c2x - a0x, e2y = c2y - a0y, e2z = c2z - a0z;
    float nx = e1y * e2z - e1z * e2y;
    float ny = e1z * e2x - e1x * e2z;
    float nz = e1x * e2y - e1y * e2x;
    const float inv_len = __builtin_amdgcn_rsqf(nx * nx + ny * ny + nz * nz);
    nx *= inv_len; ny *= inv_len; nz *= inv_len;
    const float kk = -(nx * p3x + ny * p3y + nz * p3z);

    // barycentric-test constants (surface-only); G == D in the reference
    const float Bc = a0x * b1z - a0z * b1x;
    const float Dc = a0x * b1y - a0y * b1x;
    const float Ec = a0x * c2z - a0z * c2x;
    const float Pc = a0y * c2x - a0x * c2y;           // gam/beta shared term
    const float Fc = Bc * Pc;
    const float den = Dc * (Ec * Dc + Fc);            // D*(E*G + F)
    const float inv_den = safe_rcp(den);
    const float invD    = safe_rcp(Dc);
    const float inv_ax  = safe_rcp(a0x);

    // ---------------- WMMA operands ----------------------------------------
    // A (rays, 16x4): lanes 0-15 -> {x,y}, lanes 16-31 -> {z,0}
    const int r0 = rayBase + laneLo;
    const float* __restrict__ po = o + (size_t)r0 * 3;
    const float* __restrict__ pd = d + (size_t)r0 * 3;
    v2f aO, aD;
    aO.x = laneHi ? po[2] : po[0];
    aO.y = laneHi ? 0.0f  : po[1];
    aD.x = laneHi ? pd[2] : pd[0];
    aD.y = laneHi ? 0.0f  : pd[1];
    // B (normals^T, 4x16): lanes 0-15 -> {vx,vy}, lanes 16-31 -> {vz,0}
    v2f bV;
    bV.x = laneHi ? nz   : nx;
    bV.y = laneHi ? 0.0f : ny;

    v8f zero = {0.f, 0.f, 0.f, 0.f, 0.f, 0.f, 0.f, 0.f};
    // D = A x B + 0 : vo tile and vd tile (row = i + 8*laneHi, col = laneLo)
    v8f vo = __builtin_amdgcn_wmma_f32_16x16x4_f32(
        false, aO, false, bV, (short)0, zero, false, false);
    v8f vd = __builtin_amdgcn_wmma_f32_16x16x4_f32(
        false, aD, false, bV, (short)0, zero, false, false);

    // ---------------- epilogue: 8 rows per lane -----------------------------
#pragma unroll
    for (int i = 0; i < 8; ++i) {
        const int rr = rayBase + i + 8 * laneHi;
        const float* __restrict__ ro = o + (size_t)rr * 3;
        const float* __restrict__ rd = d + (size_t)rr * 3;
        const float oxi = ro[0], oyi = ro[1], ozi = ro[2];
        const float dxi = rd[0], dyi = rd[1], dzi = rd[2];

        const float t  = -(kk + vo[i]) * __builtin_amdgcn_rcpf(vd[i]);
        const float rx = fmaf(t, dxi, oxi);
        const float ry = fmaf(t, dyi, oyi);
        const float rz = fmaf(t, dzi, ozi);

        const float A   = fmaf(a0x, rz, -(a0z * rx));
        const float C   = fmaf(a0x, ry, -(a0y * rx));
        const float num = fmaf(A, Dc, -(Bc * C));
        const float gam = Dc * num * inv_den;                 // G*(AD-BC)/safe(D(EG+F))
        const float beta  = fmaf(gam, Pc, C) * invD;
        const float alpha = (rx - fmaf(beta, b1x, gam * c2x)) * inv_ax;

        const bool  hit = (beta > 0.0f) && (gam > 0.0f) && (alpha > 0.0f);
        const float res = hit ? t : 0.0f;
        out[(size_t)rr * N_SURF + col] = res;
    }
}

extern "C" void kernel_launch(void* const* d_in, const int* in_sizes, int n_in,
                              void* d_out, int out_size, void* d_ws, size_t ws_size,
                              hipStream_t stream) {
    (void)in_sizes; (void)n_in; (void)d_ws; (void)ws_size; (void)out_size;
    const float* o = (const float*)d_in[0];
    const float* d = (const float*)d_in[1];
    const float* V = (const float*)d_in[2];
    float* out = (float*)d_out;

    // 65536 tiles (512 ray-tiles x 128 surf-tiles), 8 waves (tiles) per block
    const int blocks = (N_RAYS / 16) * (N_SURF / 16) / 8;   // 8192
    ReflectionRayTracer_kernel<<<blocks, 256, 0, stream>>>(o, d, V, out);
}